// DynamicGraphBuilder_54640573939919
// MI455X (gfx1250) — compile-verified
//
#include <hip/hip_runtime.h>
#include <hip/hip_bf16.h>

#define NPTS 4096*2            // 8192
#define EARTH_R 6371.0f
#define MAX_DIST 300.0f
#define THRESHOLD 0.3f
#define IT 32                   // rows per block
#define JT 1024                 // cols per block (256 thr * 4)

typedef float v4f __attribute__((ext_vector_type(4)));
typedef int   v4i __attribute__((vector_size(16)));   // matches builtin param type

#define AS1 __attribute__((address_space(1)))
#define AS3 __attribute__((address_space(3)))

// ---------------------------------------------------------------------------
// Pair kernel (defined first so its asm is at the top of the snippet).
// Block = 256 threads (8 wave32) owns an IT x JT tile.
//   - i-tile (IT rows * 32 B = 1 KB) staged to LDS via async global->LDS
//   - j data (4 points * float4) held in registers
//   - outputs streamed with non-temporal b128 stores
// P layout per point (8 floats, AoS):
//   [lat_rad, lon_rad, sin(lat), cos(lat), sin(wdir), cos(wdir), wspd/10, 0]
// ---------------------------------------------------------------------------
__global__ __launch_bounds__(256)
void wind_pair_kernel(const float* __restrict__ P, float* __restrict__ out) {
    __shared__ float sI[IT * 8];

    const int tid = threadIdx.x;
    const int i0  = blockIdx.y * IT;
    const int j0  = blockIdx.x * JT + tid * 4;

    // ---- stage i-tile into LDS (64 lanes x 16 B = 1 KB), async global->LDS
    if (tid < 64) {
        const float* gsrc = P + (size_t)i0 * 8 + (size_t)tid * 4;
        float*       ldst = sI + tid * 4;
#if __has_builtin(__builtin_amdgcn_global_load_async_to_lds_b128)
        __builtin_amdgcn_global_load_async_to_lds_b128(
            (AS1 v4i*)(uintptr_t)gsrc,                 // flat -> AS1: same bits
            (AS3 v4i*)(uint32_t)(uintptr_t)ldst,       // low 32 bits = LDS offset
            0, 0);
#else
        {
            uint64_t ga = (uint64_t)(uintptr_t)gsrc;
            uint32_t la = (uint32_t)(uintptr_t)ldst;
            asm volatile("global_load_async_to_lds_b128 %0, %1, off"
                         :: "v"(la), "v"(ga) : "memory");
        }
#endif
    }
#if __has_builtin(__builtin_amdgcn_s_wait_asynccnt)
    __builtin_amdgcn_s_wait_asynccnt(0);
#else
    asm volatile("s_wait_asynccnt 0" ::: "memory");
#endif
    __syncthreads();

    // ---- load this thread's 4 j-points (need first half of AoS record only)
    v4f J[4];
#pragma unroll
    for (int v = 0; v < 4; ++v)
        J[v] = *(const v4f*)(P + (size_t)(j0 + v) * 8);

    const float negRcp100 = -0.01f;
    const float twoR = 2.0f * EARTH_R;

    for (int it = 0; it < IT; ++it) {
        const int i = i0 + it;
        v4f A = *(const v4f*)(sI + it * 8);      // latr, lonr, sin(lat), cos(lat)
        v4f B = *(const v4f*)(sI + it * 8 + 4);  // sin(w), cos(w), wspd/10, -

        v4f e4, m4;
        float ev[4], mv[4];
#pragma unroll
        for (int v = 0; v < 4; ++v) {
            const int j = j0 + v;
            const float latr2 = J[v].x, lonr2 = J[v].y;
            const float sl2 = J[v].z, cl2 = J[v].w;

            const float dlat = latr2 - A.x;
            const float dlon = lonr2 - A.y;

            // haversine: one sin + one sincos
            const float sh  = __sinf(0.5f * dlat);
            float shl, chl;
            __sincosf(0.5f * dlon, &shl, &chl);
            float a = sh * sh + (A.w * cl2) * (shl * shl);
            a = fminf(fmaxf(a, 1e-12f), 1.0f - 1e-12f);
            // 2*atan2(sqrt(a), sqrt(1-a)) == 2*asin(sqrt(a))
            const float dist = twoR * asinf(__builtin_sqrtf(a));

            // double angle: sin(dlon), cos(dlon) from sincos(dlon/2)
            const float sdl = 2.0f * shl * chl;
            const float cdl = 1.0f - 2.0f * shl * shl;

            float x = sdl * cl2;
            float y = A.w * sl2 - A.z * cl2 * cdl;
            const bool diag = (i == j);
            if (diag) { x = 0.0f; y = 1.0f; }

            // cos(angle_diff_rad) == cos(atan2(x,y) - w) == (y*cw + x*sw)/r
            const float invr  = __builtin_amdgcn_rsqf(x * x + y * y);
            const float align = (y * B.y + x * B.x) * invr;

            const float infl = align * B.z * __expf(dist * negRcp100);
            const bool m = (!diag) && (dist <= MAX_DIST) && (infl > THRESHOLD);
            ev[v] = m ? infl : 0.0f;
            mv[v] = m ? 1.0f : 0.0f;
        }
        e4.x = ev[0]; e4.y = ev[1]; e4.z = ev[2]; e4.w = ev[3];
        m4.x = mv[0]; m4.y = mv[1]; m4.z = mv[2]; m4.w = mv[3];

        const size_t base = (size_t)i * NPTS + (size_t)j0;
        __builtin_nontemporal_store(e4, (v4f*)(out + base));
        __builtin_nontemporal_store(m4, (v4f*)(out + (size_t)NPTS * NPTS + base));
    }
}

// ---------------------------------------------------------------------------
// Per-point precompute. AoS layout, 8 floats per point:
//   [lat_rad, lon_rad, sin(lat), cos(lat), sin(wdir), cos(wdir), wspd/10, 0]
// ---------------------------------------------------------------------------
__global__ __launch_bounds__(256)
void wind_precompute_kernel(const float* __restrict__ pos,
                            const float* __restrict__ wspd,
                            const float* __restrict__ wdir,
                            float* __restrict__ P) {
    int t = blockIdx.x * 256 + threadIdx.x;
    if (t >= NPTS) return;
    const float D2R = 0.017453292519943295f;
    float latr = pos[2 * t] * D2R;
    float lonr = pos[2 * t + 1] * D2R;
    float sl, cl;
    __sincosf(latr, &sl, &cl);
    float w = wdir[t] * D2R;
    float sw, cw;
    __sincosf(w, &sw, &cw);
    v4f a; a.x = latr; a.y = lonr; a.z = sl; a.w = cl;
    v4f b; b.x = sw;   b.y = cw;   b.z = wspd[t] * 0.1f; b.w = 0.0f;
    v4f* Pp = (v4f*)P;
    Pp[2 * t]     = a;
    Pp[2 * t + 1] = b;
}

// ---------------------------------------------------------------------------
extern "C" void kernel_launch(void* const* d_in, const int* in_sizes, int n_in,
                              void* d_out, int out_size, void* d_ws, size_t ws_size,
                              hipStream_t stream) {
    const float* pos  = (const float*)d_in[0];   // (N,2) float32
    const float* wspd = (const float*)d_in[1];   // (N,)  float32
    const float* wdir = (const float*)d_in[2];   // (N,)  float32
    float* P   = (float*)d_ws;                   // N*8 floats = 256 KB scratch
    float* out = (float*)d_out;                  // [edge N*N | mask N*N]

    wind_precompute_kernel<<<NPTS / 256, 256, 0, stream>>>(pos, wspd, wdir, P);

    dim3 grid(NPTS / JT, NPTS / IT);             // (8, 256)
    wind_pair_kernel<<<grid, 256, 0, stream>>>(P, out);
}